// GNNLoadRegressor_13958643712827
// MI455X (gfx1250) — compile-verified
//
#include <hip/hip_runtime.h>
#include <math.h>

#define NEG_SLOPE 0.2f
#define KMAX 240          // max inner dim across layers
#define MAXW 240          // max feature width across layers

typedef __attribute__((ext_vector_type(2))) float v2f;
typedef __attribute__((ext_vector_type(8))) float v8f;

// ---------------------------------------------------------------------------
// Pack W (K x Nc, row-major) into Wt (Ncp x K), transposed + zero-padded so
// the GEMM inner loop needs no bounds checks and B loads are contiguous.
// ---------------------------------------------------------------------------
__global__ void gat_pack_wt(const float* __restrict__ W, float* __restrict__ Wt,
                            int K, int Nc, int Ncp)
{
    const int i = blockIdx.x * blockDim.x + threadIdx.x;
    if (i >= Ncp * K) return;
    const int n = i / K;
    const int k = i - n * K;
    Wt[i] = (n < Nc) ? W[(size_t)k * Nc + n] : 0.0f;
}

// ---------------------------------------------------------------------------
// GEMM: H = X @ W  via V_WMMA_F32_16X16X4_F32.
// Block = 8 waves; block owns one 16-wide N column (B panel cached in LDS)
// and 8 consecutive 16-row M tiles (one per wave). K % 4 == 0 always.
// All inner-loop loads are unguarded b64 (global for A, ds for B).
// ---------------------------------------------------------------------------
__global__ void gat_gemm_wmma(const float* __restrict__ X,
                              const float* __restrict__ Wt,   // Ncp x K, padded
                              float* __restrict__ H,
                              int Nr, int K, int Nc, int tilesN)
{
    __shared__ float ldsB[16 * KMAX];          // 15360 B max

    const int lane = threadIdx.x & 31;
    const int wv   = threadIdx.x >> 5;         // 0..7
    const int bN   = blockIdx.x % tilesN;
    const int bM   = blockIdx.x / tilesN;

    // cooperative fill: 16 rows of Wt, K floats each
    const float* wsrc = Wt + (size_t)(bN * 16) * K;
    for (int i = threadIdx.x; i < 16 * K; i += blockDim.x)
        ldsB[i] = wsrc[i];
    __syncthreads();

    const int tM = bM * 8 + wv;
    if (tM * 16 >= Nr) return;                 // wave-uniform exit

    const int mrow = tM * 16 + (lane & 15);    // Nr % 16 == 0 for this workload
    const int kh   = (lane >> 4) << 1;         // 0 or 2 within each K=4 step
    const float* arow = X    + (size_t)mrow * K + kh;
    const float* brow = ldsB + (size_t)(lane & 15) * K + kh;

    v8f acc = {};
    #pragma unroll 4
    for (int k0 = 0; k0 < K; k0 += 4) {
        const v2f a = *(const v2f*)(arow + k0);   // global_load_b64
        const v2f b = *(const v2f*)(brow + k0);   // ds_load_b64
        acc = __builtin_amdgcn_wmma_f32_16x16x4_f32(
                  /*neg_a=*/false, a, /*neg_b=*/false, b,
                  /*c_mod=*/(short)0, acc, /*reuse_a=*/false, /*reuse_b=*/false);
    }

    const int ncol = bN * 16 + (lane & 15);
    if (ncol < Nc) {
        const int rbase = tM * 16 + ((lane >> 4) << 3);   // +0 or +8
        #pragma unroll
        for (int v = 0; v < 8; ++v)
            H[(size_t)(rbase + v) * Nc + ncol] = acc[v];
    }
}

// ---------------------------------------------------------------------------
// alpha_s[n,h] = sum_c H[n,h,c]*a_src[h,c]   (and same for alpha_d)
// ---------------------------------------------------------------------------
__global__ void gat_alphas(const float* __restrict__ H,
                           const float* __restrict__ aSrcW,
                           const float* __restrict__ aDstW,
                           float* __restrict__ aS, float* __restrict__ aD,
                           int Nn, int heads, int c)
{
    const int n = blockIdx.x * blockDim.x + threadIdx.x;
    if (n >= Nn) return;
    const size_t base = (size_t)n * heads * c;
    for (int hd = 0; hd < heads; ++hd) {
        float s = 0.0f, d = 0.0f;
        for (int j = 0; j < c; ++j) {
            const float v = H[base + hd * c + j];
            s += v * aSrcW[hd * c + j];
            d += v * aDstW[hd * c + j];
        }
        aS[(size_t)n * heads + hd] = s;
        aD[(size_t)n * heads + hd] = d;
    }
}

__global__ void gat_init_mz(float* __restrict__ m, float* __restrict__ z, int total)
{
    const int i = blockIdx.x * blockDim.x + threadIdx.x;
    if (i >= total) return;
    m[i] = -INFINITY;
    z[i] = 0.0f;
}

__global__ void gat_init_zero(float* __restrict__ p, int total)
{
    const int i = blockIdx.x * blockDim.x + threadIdx.x;
    if (i < total) p[i] = 0.0f;
}

__device__ __forceinline__ void atomicMaxF(float* addr, float val)
{
    // sign-split trick: correct with -inf init, no CAS loop
    if (!(val < 0.0f)) atomicMax((int*)addr, __float_as_int(val));
    else               atomicMin((unsigned int*)addr, __float_as_uint(val));
}

__device__ __forceinline__ float leaky(float v)
{
    return v > 0.0f ? v : v * NEG_SLOPE;
}

// one thread per edge; loops heads
__global__ void gat_edge_max(const int* __restrict__ srcI, const int* __restrict__ dstI,
                             int Ereal, int ET,
                             const float* __restrict__ aS, const float* __restrict__ aD,
                             float* __restrict__ m, int heads)
{
    const int e = blockIdx.x * blockDim.x + threadIdx.x;
    if (e >= ET) return;
    int s, d;
    if (e < Ereal) { s = srcI[e]; d = dstI[e]; } else { s = d = e - Ereal; }
    for (int hd = 0; hd < heads; ++hd) {
        const float v = leaky(aS[(size_t)s * heads + hd] + aD[(size_t)d * heads + hd]);
        atomicMaxF(&m[(size_t)d * heads + hd], v);
    }
}

__global__ void gat_edge_sum(const int* __restrict__ srcI, const int* __restrict__ dstI,
                             int Ereal, int ET,
                             const float* __restrict__ aS, const float* __restrict__ aD,
                             const float* __restrict__ m, float* __restrict__ z, int heads)
{
    const int e = blockIdx.x * blockDim.x + threadIdx.x;
    if (e >= ET) return;
    int s, d;
    if (e < Ereal) { s = srcI[e]; d = dstI[e]; } else { s = d = e - Ereal; }
    for (int hd = 0; hd < heads; ++hd) {
        const float v  = leaky(aS[(size_t)s * heads + hd] + aD[(size_t)d * heads + hd]);
        const float ex = __expf(v - m[(size_t)d * heads + hd]);
        atomicAdd(&z[(size_t)d * heads + hd], ex);
    }
}

// one thread per (edge, head); loops c, atomic-adds weighted feature
__global__ void gat_edge_scatter(const int* __restrict__ srcI, const int* __restrict__ dstI,
                                 int Ereal, int ET,
                                 const float* __restrict__ aS, const float* __restrict__ aD,
                                 const float* __restrict__ m, const float* __restrict__ z,
                                 const float* __restrict__ H, float* __restrict__ acc,
                                 int heads, int c)
{
    const int t = blockIdx.x * blockDim.x + threadIdx.x;
    const int total = ET * heads;
    if (t >= total) return;
    const int e  = t / heads;
    const int hd = t - e * heads;
    int s, d;
    if (e < Ereal) { s = srcI[e]; d = dstI[e]; } else { s = d = e - Ereal; }

    const float v    = leaky(aS[(size_t)s * heads + hd] + aD[(size_t)d * heads + hd]);
    const float ex   = __expf(v - m[(size_t)d * heads + hd]);
    const float attn = ex / z[(size_t)d * heads + hd];

    const float* hrow = H   + (size_t)s * heads * c + (size_t)hd * c;
    float*       arow = acc + (size_t)d * heads * c + (size_t)hd * c;
    __builtin_prefetch(hrow, 0, 0);            // global_prefetch_b8 on gfx1250
    for (int j = 0; j < c; ++j)
        atomicAdd(&arow[j], hrow[j] * attn);
}

// out = relu(acc + b)
__global__ void gat_finalize(float* __restrict__ acc, const float* __restrict__ b,
                             int total, int hc)
{
    const int i = blockIdx.x * blockDim.x + threadIdx.x;
    if (i >= total) return;
    const float v = acc[i] + b[i % hc];
    acc[i] = v > 0.0f ? v : 0.0f;
}

// ---------------------------------------------------------------------------
extern "C" void kernel_launch(void* const* d_in, const int* in_sizes, int n_in,
                              void* d_out, int out_size, void* d_ws, size_t ws_size,
                              hipStream_t stream)
{
    const float* x0  = (const float*)d_in[0];
    const int*   ei  = (const int*)d_in[1];
    const int N  = in_sizes[0] / 32;
    const int E  = in_sizes[1] / 2;
    const int ET = E + N;                      // + self loops
    const int* srcIdx = ei;
    const int* dstIdx = ei + E;

    static const int din[5]    = {32, 240, 120, 48, 24};
    static const int cDim[5]   = {24, 24, 24, 24, 12};
    static const int hHeads[5] = {10, 5, 2, 1, 1};

    float* ws   = (float*)d_ws;
    float* bufH = ws;                               // N x 240  (h = x@W)
    float* bufX = ws + (size_t)N * MAXW;            // N x 240  (layer output / next input)
    float* aS   = ws + 2 * (size_t)N * MAXW;        // N x 16
    float* aD   = aS + (size_t)N * 16;              // N x 16
    float* mB   = aD + (size_t)N * 16;              // N x 16
    float* zB   = mB + (size_t)N * 16;              // N x 16
    float* Wt   = zB + (size_t)N * 16;              // <= 240 x 240 packed weights

    const float* xin = x0;
    for (int L = 0; L < 5; ++L) {
        const int K  = din[L];
        const int c  = cDim[L];
        const int hh = hHeads[L];
        const int hc = c * hh;
        const float* W   = (const float*)d_in[3 + 4 * L];
        const float* b   = (const float*)d_in[4 + 4 * L];
        const float* as_ = (const float*)d_in[5 + 4 * L];
        const float* ad_ = (const float*)d_in[6 + 4 * L];
        float* acc = (L == 4) ? (float*)d_out : bufX;

        // 0) pack + transpose + pad weights
        const int tilesN = (hc + 15) / 16;
        const int Ncp    = tilesN * 16;
        gat_pack_wt<<<(Ncp * K + 255) / 256, 256, 0, stream>>>(W, Wt, K, hc, Ncp);

        // 1) GEMM via WMMA: bufH = xin @ W
        const int tilesM  = (N + 15) / 16;
        const int blocksM = (tilesM + 7) / 8;       // 8 M-tiles per block
        gat_gemm_wmma<<<blocksM * tilesN, 256, 0, stream>>>(xin, Wt, bufH, N, K, hc, tilesN);

        // 2) attention logits per node/head
        gat_alphas<<<(N + 255) / 256, 256, 0, stream>>>(bufH, as_, ad_, aS, aD, N, hh, c);

        // 3) init reductions (acc may alias xin; GEMM above already consumed it)
        gat_init_mz<<<(N * hh + 255) / 256, 256, 0, stream>>>(mB, zB, N * hh);
        gat_init_zero<<<(N * hc + 255) / 256, 256, 0, stream>>>(acc, N * hc);

        // 4) softmax-max, 5) softmax-sum, 6) weighted scatter
        gat_edge_max<<<(ET + 255) / 256, 256, 0, stream>>>(srcIdx, dstIdx, E, ET, aS, aD, mB, hh);
        gat_edge_sum<<<(ET + 255) / 256, 256, 0, stream>>>(srcIdx, dstIdx, E, ET, aS, aD, mB, zB, hh);
        const int totEH = ET * hh;
        gat_edge_scatter<<<(totEH + 255) / 256, 256, 0, stream>>>(srcIdx, dstIdx, E, ET,
                                                                  aS, aD, mB, zB, bufH, acc, hh, c);

        // 7) bias + ReLU in place
        gat_finalize<<<(N * hc + 255) / 256, 256, 0, stream>>>(acc, b, N * hc, hc);

        xin = acc;
    }
    (void)n_in; (void)out_size; (void)ws_size;
}